// Scatter_24068996727295
// MI455X (gfx1250) — compile-verified
//
#include <hip/hip_runtime.h>

#define AS1 __attribute__((address_space(1)))
#define AS3 __attribute__((address_space(3)))

typedef __attribute__((ext_vector_type(4))) float v4f;
typedef int v4i __attribute__((vector_size(16)));  // matches builtin param type

#if defined(__has_builtin)
#if __has_builtin(__builtin_amdgcn_global_load_async_to_lds_b128)
#define HAVE_ASYNC_LDS 1
#endif
#if __has_builtin(__builtin_amdgcn_s_wait_asynccnt)
#define HAVE_WAIT_ASYNC 1
#endif
#endif

#define CHUNK 256            // edges per wave
#define TILE 32              // edges per async tile (32 rows * 128B = 4KB)
#define WAVES_PER_BLOCK 8
#define BLOCK (WAVES_PER_BLOCK * 32)

// Temporal hint: non-temporal load (src is streamed exactly once; keep L2 for
// the index stream and the atomic-accumulated output).
#define CPOL_NT 1

__device__ __forceinline__ void waitAsyncLe8() {
#ifdef HAVE_WAIT_ASYNC
  __builtin_amdgcn_s_wait_asynccnt(8);
#else
  asm volatile("s_wait_asynccnt 8" ::: "memory");
#endif
}

__device__ __forceinline__ void waitAsync0() {
#ifdef HAVE_WAIT_ASYNC
  __builtin_amdgcn_s_wait_asynccnt(0);
#else
  asm volatile("s_wait_asynccnt 0" ::: "memory");
#endif
}

// Copy one 32-row x 128B tile (4KB) global -> LDS.
// Each of the 8 issues moves 512B (32 lanes x 16B), contiguous. The 512B
// stepping uses the instruction's 24-bit immediate offset (applied to both
// the global and LDS addresses), so only one address VGPR pair is needed.
__device__ __forceinline__ void tileLoad(const float* gRow0, AS3 float* lBase, int lane) {
#ifdef HAVE_ASYNC_LDS
  AS1 v4i* g = (AS1 v4i*)((const char*)gRow0 + lane * 16);
  AS3 v4i* l = (AS3 v4i*)((AS3 char*)lBase + lane * 16);
#define ASYNC_ISSUE(K) \
  __builtin_amdgcn_global_load_async_to_lds_b128(g, l, (K) * 512, CPOL_NT)
  ASYNC_ISSUE(0); ASYNC_ISSUE(1); ASYNC_ISSUE(2); ASYNC_ISSUE(3);
  ASYNC_ISSUE(4); ASYNC_ISSUE(5); ASYNC_ISSUE(6); ASYNC_ISSUE(7);
#undef ASYNC_ISSUE
#else
  // Synchronous fallback: plain vector load + ds store.
  const v4f* g = (const v4f*)((const char*)gRow0 + lane * 16);
  AS3 v4f* l = (AS3 v4f*)((AS3 char*)lBase + lane * 16);
#pragma unroll
  for (int k = 0; k < 8; ++k) l[k * 32] = g[k * 32];
#endif
}

__global__ __launch_bounds__(BLOCK) void seg_sum_kernel(
    const float* __restrict__ src, const int* __restrict__ idx,
    float* __restrict__ out, int E) {
  __shared__ float smemRaw[WAVES_PER_BLOCK * 2 * TILE * 32];  // 64KB
  AS3 float* smem = (AS3 float*)smemRaw;

  const int lane = threadIdx.x & 31;          // lane == channel (32 f32/row)
  const int slot = threadIdx.x >> 5;
  const int wave = blockIdx.x * WAVES_PER_BLOCK + slot;
  const int e0 = wave * CHUNK;                // < 3.2M, fits in int
  if (e0 >= E) return;                        // whole-wave uniform exit
  const int e1 = (e0 + CHUNK < E) ? (e0 + CHUNK) : E;
  const int nEdges = e1 - e0;
  const int nTiles = nEdges >> 5;
  const int tail = nEdges & 31;

  AS3 float* myLds = smem + slot * (2 * TILE * 32);

  // Prime the double buffer.
  if (nTiles > 0) tileLoad(src + (size_t)e0 * 32, myLds, lane);
  if (nTiles > 1) tileLoad(src + (size_t)(e0 + TILE) * 32, myLds + TILE * 32, lane);

  int cur = idx[e0];   // first segment id of this chunk (uniform)
  float acc = 0.f;

  for (int t = 0; t < nTiles; ++t) {
    const int tb = e0 + t * TILE;
    const int myIdx = idx[tb + lane];         // coalesced: 32 indices per wave

    if (t < nTiles - 1) waitAsyncLe8();       // tile t complete, t+1 may fly
    else                waitAsync0();         // last tile: drain

    AS3 float* buf = myLds + (t & 1) * (TILE * 32);
#pragma unroll 8
    for (int j = 0; j < TILE; ++j) {
      const int seg = __builtin_amdgcn_readlane(myIdx, j);  // uniform seg id
      const float v = buf[j * 32 + lane];                   // conflict-free LDS
      if (seg != cur) {                                     // scalar branch
        atomicAdd(out + (size_t)cur * 32 + lane, acc);
        acc = 0.f;
        cur = seg;
      }
      acc += v;
    }

    // Prefetch tile t+2 into the buffer we just drained.
    if (t + 2 < nTiles)
      tileLoad(src + (size_t)(tb + 2 * TILE) * 32, myLds + (t & 1) * (TILE * 32), lane);
  }

  // Tail (<32 edges) straight from global (E=3.2M is tile-divisible; kept for safety).
  if (tail) {
    const int tb = e0 + nTiles * TILE;
    const int myIdx = (tb + lane < e1) ? idx[tb + lane] : cur;
    for (int j = 0; j < tail; ++j) {
      const int seg = __builtin_amdgcn_readlane(myIdx, j);
      const float v = src[(size_t)(tb + j) * 32 + lane];
      if (seg != cur) {
        atomicAdd(out + (size_t)cur * 32 + lane, acc);
        acc = 0.f;
        cur = seg;
      }
      acc += v;
    }
  }

  // Final flush for this chunk's last (possibly shared) segment.
  atomicAdd(out + (size_t)cur * 32 + lane, acc);
}

__global__ void zero_kernel(float* out, int n4) {
  int i = blockIdx.x * blockDim.x + threadIdx.x;
  if (i < n4) *(v4f*)(out + (size_t)i * 4) = (v4f){0.f, 0.f, 0.f, 0.f};
}

extern "C" void kernel_launch(void* const* d_in, const int* in_sizes, int n_in,
                              void* d_out, int out_size, void* d_ws, size_t ws_size,
                              hipStream_t stream) {
  const float* src = (const float*)d_in[0];
  const int* idx = (const int*)d_in[1];
  // d_in[2] = pos, only defines N; unused on device.
  float* out = (float*)d_out;

  const int E = in_sizes[0] / 32;  // src is [E, 8, 4]
  const int n = out_size;          // N * 32, divisible by 4

  {
    int n4 = n / 4;
    int blocks = (n4 + 255) / 256;
    if (blocks > 0) zero_kernel<<<blocks, 256, 0, stream>>>(out, n4);
  }
  {
    int numWaves = (E + CHUNK - 1) / CHUNK;
    int blocks = (numWaves + WAVES_PER_BLOCK - 1) / WAVES_PER_BLOCK;
    if (blocks > 0) seg_sum_kernel<<<blocks, BLOCK, 0, stream>>>(src, idx, out, E);
  }
}